// GCCF_Encoder_41618233098461
// MI455X (gfx1250) — compile-verified
//
#include <hip/hip_runtime.h>
#include <stdint.h>

// ---------------- problem constants (match reference) ----------------
#define N_USERS_C   100000
#define N_ITEMS_C   50000
#define N_NODES_C   150000
#define EMB_C       64
#define NODE_FLOATS (N_NODES_C * EMB_C)   // 9,600,000
#define USER_FLOATS (N_USERS_C * EMB_C)   // 6,400,000
#define N_LAYERS    3

#define THREADS 256
#define CHUNK   1024   // edges staged into LDS per block (12 KB LDS)

// ---------------- CDNA5 async global->LDS staging ----------------
// Probe-confirmed signatures on this toolchain:
//   b32 : (global int*,  shared int*,  i32 offset, i32 cpol)
//   b128: (global v4i*,  shared v4i*,  i32 offset, i32 cpol)   [v4i = vector_size(16) int]
#if __has_builtin(__builtin_amdgcn_global_load_async_to_lds_b32) && \
    __has_builtin(__builtin_amdgcn_s_wait_asynccnt)
#define USE_ASYNC 1
#else
#define USE_ASYNC 0
#endif

#if USE_ASYNC
#define AS_GLOBAL __attribute__((address_space(1)))
#define AS_SHARED __attribute__((address_space(3)))
typedef int v4i_async __attribute__((vector_size(16)));

__device__ __forceinline__ void stage_b32(const void* g, void* l) {
    __builtin_amdgcn_global_load_async_to_lds_b32(
        (AS_GLOBAL int*)g, (AS_SHARED int*)l, 0, 0);
}
#if __has_builtin(__builtin_amdgcn_global_load_async_to_lds_b128)
#define HAVE_ASYNC_B128 1
__device__ __forceinline__ void stage_b128(const void* g, void* l) {
    __builtin_amdgcn_global_load_async_to_lds_b128(
        (AS_GLOBAL v4i_async*)g, (AS_SHARED v4i_async*)l, 0, 0);
}
#else
#define HAVE_ASYNC_B128 0
#endif
#endif // USE_ASYNC

// ---------------- codegen probe (never launched) ----------------
// First function in the file -> its lowering appears at the top of the
// disasm snippet. Verifies that the relaxed agent-scope f32 fetch_add used
// by the scatter kernel lowers to hardware global_atomic_add_f32 (no CAS loop).
__global__ void probe_atomic_kernel(float* p, float v) {
    __hip_atomic_fetch_add(p + threadIdx.x, v, __ATOMIC_RELAXED, __HIP_MEMORY_SCOPE_AGENT);
}

// ---------------- kernels ----------------

// COO SpMM scatter: y[row] += val * x[col].
// One block = one CHUNK of edges (grid covers all chunks exactly).
// Stage (val,row,col) into LDS via async b128 DMA, then each of the 8 waves
// processes CHUNK/8 edges; one wave covers the full 64-float embedding row
// (lane -> 2 floats, one coalesced 256B transaction).
__global__ void spmm_scatter_kernel(const float* __restrict__ val,
                                    const int*   __restrict__ row,
                                    const int*   __restrict__ col,
                                    const float* __restrict__ x,
                                    float*       __restrict__ y,
                                    int n_edges) {
    __shared__ __align__(16) float s_val[CHUNK];
    __shared__ __align__(16) int   s_row[CHUNK];
    __shared__ __align__(16) int   s_col[CHUNK];

    const int base = blockIdx.x * CHUNK;
    const int cnt  = min(CHUNK, n_edges - base);
    const int t    = threadIdx.x;

#if USE_ASYNC && HAVE_ASYNC_B128
    // 16B async DMA per thread per array; cnt4 <= 256 == THREADS so one shot.
    // base is a multiple of CHUNK -> global addresses 16B aligned.
    const int cnt4 = cnt >> 2;
    if (t < cnt4) {
        stage_b128(val + base + 4 * t, &s_val[4 * t]);
        stage_b128(row + base + 4 * t, &s_row[4 * t]);
        stage_b128(col + base + 4 * t, &s_col[4 * t]);
    }
    const int tt = (cnt4 << 2) + t;      // <=3 element scalar tail
    if (tt < cnt) {
        s_val[tt] = val[base + tt];
        s_row[tt] = row[base + tt];
        s_col[tt] = col[base + tt];
    }
    __builtin_amdgcn_s_wait_asynccnt(0);
#elif USE_ASYNC
    for (int i = t; i < cnt; i += THREADS) {
        stage_b32(val + base + i, &s_val[i]);
        stage_b32(row + base + i, &s_row[i]);
        stage_b32(col + base + i, &s_col[i]);
    }
    __builtin_amdgcn_s_wait_asynccnt(0);
#else
    for (int i = t; i < cnt; i += THREADS) {
        s_val[i] = val[base + i];
        s_row[i] = row[base + i];
        s_col[i] = col[base + i];
    }
#endif
    __syncthreads();

    const int lane = t & 31;
    const int wv   = t >> 5;                    // wave index in block: 0..7
    const int per  = CHUNK / (THREADS / 32);    // 128 edges per wave
    const int jb   = wv * per;
    const int je   = min(jb + per, cnt);

    #pragma unroll 2
    for (int j = jb; j < je; ++j) {
        const float v = s_val[j];               // LDS broadcast reads
        const int   r = s_row[j];
        const int   c = s_col[j];
        const float2 xv = *(const float2*)(x + (size_t)c * EMB_C + 2 * lane);
        float* dst = y + (size_t)r * EMB_C + 2 * lane;
        // no-return agent-scope f32 atomics -> global_atomic_add_f32 (STOREcnt)
        __hip_atomic_fetch_add(dst,     v * xv.x, __ATOMIC_RELAXED, __HIP_MEMORY_SCOPE_AGENT);
        __hip_atomic_fetch_add(dst + 1, v * xv.y, __ATOMIC_RELAXED, __HIP_MEMORY_SCOPE_AGENT);
    }
}

// One float4 pass: x = concat(user,item); out = x; y = 0.
__global__ void init_kernel(const float4* __restrict__ u4,
                            const float4* __restrict__ it4,
                            float4* __restrict__ x4,
                            float4* __restrict__ y4,
                            float4* __restrict__ out4) {
    const int i = blockIdx.x * blockDim.x + threadIdx.x;
    if (i >= NODE_FLOATS / 4) return;
    const float4 v = (i < USER_FLOATS / 4) ? u4[i] : it4[i - USER_FLOATS / 4];
    x4[i]   = v;
    out4[i] = v;
    y4[i]   = make_float4(0.f, 0.f, 0.f, 0.f);
}

// x = relu(y); out += x; y = 0; final layer folds the /(N_LAYERS+1) scale.
__global__ void relu_acc_kernel(float4* __restrict__ y4,
                                float4* __restrict__ x4,
                                float4* __restrict__ out4,
                                int is_final) {
    const int i = blockIdx.x * blockDim.x + threadIdx.x;
    if (i >= NODE_FLOATS / 4) return;
    float4 v = y4[i];
    v.x = fmaxf(v.x, 0.f);
    v.y = fmaxf(v.y, 0.f);
    v.z = fmaxf(v.z, 0.f);
    v.w = fmaxf(v.w, 0.f);
    y4[i] = make_float4(0.f, 0.f, 0.f, 0.f);
    x4[i] = v;
    float4 o = out4[i];
    o.x += v.x; o.y += v.y; o.z += v.z; o.w += v.w;
    if (is_final) {
        const float s = 1.0f / (N_LAYERS + 1);
        o.x *= s; o.y *= s; o.z *= s; o.w *= s;
    }
    out4[i] = o;
}

// ---------------- host entry ----------------
extern "C" void kernel_launch(void* const* d_in, const int* in_sizes, int n_in,
                              void* d_out, int out_size, void* d_ws, size_t ws_size,
                              hipStream_t stream) {
    const float* user_emb = (const float*)d_in[0];
    const float* item_emb = (const float*)d_in[1];
    const float* adj_val  = (const float*)d_in[2];
    const int*   adj_row  = (const int*)d_in[3];
    const int*   adj_col  = (const int*)d_in[4];
    const int    n_edges  = in_sizes[2];

    float* x   = (float*)d_ws;            // [N_NODES, 64] current layer input
    float* y   = x + NODE_FLOATS;         // [N_NODES, 64] scatter accumulator
    float* out = (float*)d_out;           // [N_NODES, 64] running layer-sum -> mean

    const int vec_n      = NODE_FLOATS / 4;
    const int vec_blocks = (vec_n + THREADS - 1) / THREADS;
    const int sc_blocks  = (n_edges + CHUNK - 1) / CHUNK;

    init_kernel<<<vec_blocks, THREADS, 0, stream>>>(
        (const float4*)user_emb, (const float4*)item_emb,
        (float4*)x, (float4*)y, (float4*)out);

    for (int layer = 0; layer < N_LAYERS; ++layer) {
        spmm_scatter_kernel<<<sc_blocks, THREADS, 0, stream>>>(
            adj_val, adj_row, adj_col, x, y, n_edges);
        relu_acc_kernel<<<vec_blocks, THREADS, 0, stream>>>(
            (float4*)y, (float4*)x, (float4*)out, layer == N_LAYERS - 1 ? 1 : 0);
    }
}